// PositionalAttention_81003083203164
// MI455X (gfx1250) — compile-verified
//
#include <hip/hip_runtime.h>
#include <hip/hip_bf16.h>
#include <cstddef>

// Problem constants (match reference)
#define BB   16
#define SS   1024
#define DD   1024
#define PE   20
#define GD   80          // 4*PE gates
#define BS   (BB*SS)     // 16384 rows

typedef float v2f __attribute__((ext_vector_type(2)));
typedef float v8f __attribute__((ext_vector_type(8)));

__device__ __forceinline__ float fsig(float x) {
  return 1.0f / (1.0f + __expf(-x));
}

// ---------------------------------------------------------------------------
// Kernel A: xg[row, g] = x[row,:] @ W_ih[g,:] + (b_ih[g]+b_hh[g])
// f32 WMMA 16x16x4. Workgroup = 8 waves = 8 M-tiles (128 rows), 5 N-tiles.
// ---------------------------------------------------------------------------
#define KC        32
#define XS_STRIDE 36     // 144B rows: 16B aligned, banks 4m.. -> conflict-free
#define WT_STRIDE 81     // odd stride -> conflict-free transpose writes

__global__ __launch_bounds__(256) void xg_gemm(
    const float* __restrict__ x, const float* __restrict__ W_ih,
    const float* __restrict__ b_ih, const float* __restrict__ b_hh,
    float* __restrict__ xg) {
  __shared__ float xs[128 * XS_STRIDE];
  __shared__ float wt[KC * WT_STRIDE];
  __shared__ float bias[GD];

  const int tid = threadIdx.x;
  const int m0  = blockIdx.x * 128;
  if (tid < GD) bias[tid] = b_ih[tid] + b_hh[tid];

  const int wv = tid >> 5, lane = tid & 31;
  const int ml = lane & 15, kh = lane >> 4;

  v8f acc[5];
  #pragma unroll
  for (int t = 0; t < 5; ++t)
    #pragma unroll
    for (int r = 0; r < 8; ++r) acc[t][r] = 0.0f;

  for (int k0 = 0; k0 < DD; k0 += KC) {
    __syncthreads();  // protect previous iteration's LDS reads
    {   // stage x[m0..m0+127, k0..k0+31] (coalesced float4)
      const int r = tid >> 1, c = (tid & 1) * 16;
      const float4* src = (const float4*)(x + (size_t)(m0 + r) * DD + k0 + c);
      float4* dst = (float4*)(xs + r * XS_STRIDE + c);
      #pragma unroll
      for (int q = 0; q < 4; ++q) dst[q] = src[q];
    }
    // stage W_ih[0..79, k0..k0+31] transposed into wt[k][n]
    for (int f = tid; f < GD * KC; f += 256) {
      const int n = f >> 5, k = f & 31;
      wt[k * WT_STRIDE + n] = W_ih[(size_t)n * DD + k0 + k];
    }
    __syncthreads();

    #pragma unroll
    for (int k = 0; k < KC; k += 4) {
      // A-tile (16x4): lane ml holds row ml, K pair selected by kh
      v2f a = *(const v2f*)(xs + (wv * 16 + ml) * XS_STRIDE + k + kh * 2);
      #pragma unroll
      for (int nt = 0; nt < 5; ++nt) {
        const int n = nt * 16 + ml;
        v2f bv;
        bv.x = wt[(k + kh)     * WT_STRIDE + n];   // K rows k0+k .. +3
        bv.y = wt[(k + kh + 2) * WT_STRIDE + n];
        acc[nt] = __builtin_amdgcn_wmma_f32_16x16x4_f32(
            false, a, false, bv, (short)0, acc[nt], false, false);
      }
    }
  }
  // epilogue: C layout -> VGPR r : lanes0-15 M=r, lanes16-31 M=8+r
  #pragma unroll
  for (int nt = 0; nt < 5; ++nt)
    #pragma unroll
    for (int r = 0; r < 8; ++r) {
      const int row = m0 + wv * 16 + r + kh * 8;
      const int g   = nt * 16 + ml;
      xg[(size_t)row * GD + g] = acc[nt][r] + bias[g];
    }
}

// ---------------------------------------------------------------------------
// Kernel B: sequential LSTM, one wave32 per batch. h broadcast via readlane.
// ---------------------------------------------------------------------------
__global__ __launch_bounds__(32) void lstm_scan(
    const float* __restrict__ xg, const float* __restrict__ W_hh,
    float* __restrict__ h_out) {
  const int b = blockIdx.x;
  const int j = threadIdx.x;
  const bool act = j < PE;
  const int jc = act ? j : 0;

  float wi[PE], wf[PE], wg[PE], wo[PE];
  #pragma unroll
  for (int k = 0; k < PE; ++k) {
    wi[k] = W_hh[(size_t)(0 * PE + jc) * PE + k];
    wf[k] = W_hh[(size_t)(1 * PE + jc) * PE + k];
    wg[k] = W_hh[(size_t)(2 * PE + jc) * PE + k];
    wo[k] = W_hh[(size_t)(3 * PE + jc) * PE + k];
  }
  float h = 0.0f, c = 0.0f;
  for (int t = 0; t < SS; ++t) {
    const float* g = xg + ((size_t)b * SS + t) * GD;
    float ai = g[jc], af = g[PE + jc], ag = g[2 * PE + jc], ao = g[3 * PE + jc];
    #pragma unroll
    for (int k = 0; k < PE; ++k) {       // constant lane -> v_readlane + SGPR fma
      const float hk = __shfl(h, k, 32);
      ai += hk * wi[k]; af += hk * wf[k];
      ag += hk * wg[k]; ao += hk * wo[k];
    }
    const float ig = fsig(ai), fg = fsig(af), og = fsig(ao);
    const float gg = tanhf(ag);
    c = fg * c + ig * gg;
    const float hn = og * tanhf(c);
    h = act ? hn : 0.0f;
    if (act) h_out[((size_t)b * SS + t) * PE + j] = h;
  }
}

// ---------------------------------------------------------------------------
// Kernel B2: heads. a=relu(mu0), c=(relu(mu1)+relu(mu2)*(t+1))/S, sigma.
// ---------------------------------------------------------------------------
__global__ __launch_bounds__(256) void heads(
    const float* __restrict__ h_ws, const float* __restrict__ W_mu,
    const float* __restrict__ b_mu, const float* __restrict__ W_sig,
    const float* __restrict__ b_sig, float* __restrict__ ac,
    float* __restrict__ sig) {
  const int idx = blockIdx.x * 256 + threadIdx.x;   // (b*S + t)
  const int t = idx & (SS - 1);
  const float* hp = h_ws + (size_t)idx * PE;
  float m0 = b_mu[0], m1 = b_mu[1], m2 = b_mu[2], sv = b_sig[0];
  #pragma unroll
  for (int k = 0; k < PE; ++k) {
    const float hk = hp[k];
    m0 += hk * W_mu[0 * PE + k];
    m1 += hk * W_mu[1 * PE + k];
    m2 += hk * W_mu[2 * PE + k];
    sv += hk * W_sig[k];
  }
  const float a  = fmaxf(m0, 0.0f);
  const float w1 = fmaxf(m1, 0.0f);
  const float w2 = fmaxf(m2, 0.0f);
  ac[2 * idx]     = a;
  ac[2 * idx + 1] = (w1 + w2 * (float)(t + 1)) * (1.0f / (float)SS);
  sig[idx] = fsig(sv);
}

// ---------------------------------------------------------------------------
// Kernel B3: mu_t = a_t*mu_{t-1} + c_t as parallel scan. One wave per batch,
// 32 elements per lane serial + 5-step shuffle composition scan across lanes.
// ---------------------------------------------------------------------------
__global__ __launch_bounds__(512) void mu_scan(
    const float* __restrict__ ac, float* __restrict__ mu) {
  const int b = threadIdx.x >> 5;
  const int lane = threadIdx.x & 31;
  const float* p = ac + ((size_t)b * SS + lane * 32) * 2;
  float as[32], cs[32];
  float A = 1.0f, C = 0.0f;
  #pragma unroll
  for (int s = 0; s < 32; ++s) {
    const float a = p[2 * s], c = p[2 * s + 1];
    as[s] = a; cs[s] = c;
    A = a * A;
    C = a * C + c;
  }
  // inclusive scan of affine composition across lanes
  #pragma unroll
  for (int d = 1; d < 32; d <<= 1) {
    const float Ap = __shfl_up(A, d, 32);
    const float Cp = __shfl_up(C, d, 32);
    if (lane >= d) { C = A * Cp + C; A = A * Ap; }
  }
  float m = __shfl_up(C, 1, 32);
  if (lane == 0) m = 0.0f;          // mu starts at 0
  float* o = mu + (size_t)b * SS + lane * 32;
  #pragma unroll
  for (int s = 0; s < 32; ++s) {
    m = as[s] * m + cs[s];
    o[s] = m;
  }
}

// ---------------------------------------------------------------------------
// Kernel C: w[j,i] = exp(-(i/(j+1)-mu)^2/(2 sigma^2)) for i<=j, then
// context = (w/||w||_2) @ x via f32 WMMA.
// One WG per (b, 32-row j-block) = 2 M-tiles, so each B register pair feeds
// two WMMAs (halves L2 traffic and vmem-per-wmma). L2-norm folded into the
// output row scale. x prefetched 32 K-rows ahead (global_prefetch_b8).
// ---------------------------------------------------------------------------
#define CCHUNK 256
#define WST    260   // 1040B rows: 8B aligned; per-tile A rows hit banks 4*ml
#define MROWS  32

__global__ __launch_bounds__(256) void attn_context(
    const float* __restrict__ x, const float* __restrict__ mu,
    const float* __restrict__ sig, float* __restrict__ out) {
  __shared__ float wt[MROWS * WST];
  __shared__ float norms[MROWS], mus[MROWS], q2[MROWS], ivj[MROWS], scl[MROWS];

  const int jt = blockIdx.x, b = blockIdx.y;   // jt: 32-row block index
  const int tid = threadIdx.x;
  if (tid < MROWS) {
    const int j = jt * MROWS + tid;
    const float m = mu[b * SS + j];
    const float s = sig[b * SS + j];
    mus[tid] = m;
    q2[tid]  = 1.0f / (2.0f * s * s);
    ivj[tid] = 1.0f / (float)(j + 1);
    norms[tid] = 0.0f;
  }
  const int Kmax = jt * MROWS + MROWS;       // triangular mask: i <= j
  const int wv = tid >> 5, lane = tid & 31;
  const int ml = lane & 15, kh = lane >> 4;
  const int r = tid & 31, ci = tid >> 5;     // phase-1 mapping: 8 thr/row

  v8f acc[2][8];
  #pragma unroll
  for (int mt = 0; mt < 2; ++mt)
    #pragma unroll
    for (int t = 0; t < 8; ++t)
      #pragma unroll
      for (int rr = 0; rr < 8; ++rr) acc[mt][t][rr] = 0.0f;

  __syncthreads();
  const float mur = mus[r], qr = q2[r], ivr = ivj[r];
  const int jrow = jt * MROWS + r;

  for (int i0 = 0; i0 < Kmax; i0 += CCHUNK) {
    const int cur = min(CCHUNK, Kmax - i0);
    // phase 1: Gaussian weights into LDS + row sum-of-squares
    float ssq = 0.0f;
    for (int ii = ci; ii < cur; ii += 8) {
      const int i = i0 + ii;
      float v = 0.0f;
      if (i <= jrow) {
        const float d = (float)i * ivr - mur;
        v = __expf(-d * d * qr);
      }
      wt[r * WST + ii] = v;
      ssq += v * v;
    }
    atomicAdd(&norms[r], ssq);            // ds_add_f32
    __syncthreads();

    // phase 2: WMMA over this K chunk; 8 N-tiles x 2 M-tiles per wave
    for (int k = 0; k < cur; k += 4) {
      v2f a0 = *(const v2f*)(wt + ml * WST + k + kh * 2);
      v2f a1 = *(const v2f*)(wt + (16 + ml) * WST + k + kh * 2);
      const int ib = i0 + k + kh;
      const float* xb = x + ((size_t)b * SS + ib) * DD;
      // prefetch x 32 K-rows ahead: lanes cover the wave's 512B N-span
      // across rows ib+32..ib+35 (two instrs, rows split by kh)
      const int ipf = ib + 32;
      if (ipf + 3 < SS) {
        const float* pf = x + ((size_t)b * SS + ipf) * DD + wv * 128 + ml * 8;
        __builtin_prefetch(pf, 0, 3);
        __builtin_prefetch(pf + 2 * DD, 0, 3);
      }
      #pragma unroll
      for (int t = 0; t < 8; ++t) {
        const int n = (wv * 8 + t) * 16 + ml;
        v2f bv;
        bv.x = xb[n];              // K row ib
        bv.y = xb[2 * DD + n];     // K row ib+2
        acc[0][t] = __builtin_amdgcn_wmma_f32_16x16x4_f32(
            false, a0, false, bv, (short)0, acc[0][t], false, false);
        acc[1][t] = __builtin_amdgcn_wmma_f32_16x16x4_f32(
            false, a1, false, bv, (short)0, acc[1][t], false, false);
      }
    }
    __syncthreads();
  }

  if (tid < MROWS)
    scl[tid] = 1.0f / fmaxf(__builtin_sqrtf(norms[tid]), 1e-12f);
  __syncthreads();

  #pragma unroll
  for (int mt = 0; mt < 2; ++mt)
    #pragma unroll
    for (int t = 0; t < 8; ++t) {
      const int n = (wv * 8 + t) * 16 + ml;
      #pragma unroll
      for (int rr = 0; rr < 8; ++rr) {
        const int M = mt * 16 + rr + kh * 8;
        const int j = jt * MROWS + M;
        out[((size_t)b * SS + j) * DD + n] = acc[mt][t][rr] * scl[M];
      }
    }
}

// ---------------------------------------------------------------------------
// Workspace layout (floats):
//  xg   : BS*80   = 1,310,720
//  h    : BS*20   =   327,680
//  (a,c): BS*2    =    32,768
//  sigma: BS      =    16,384
//  mu   : BS      =    16,384           total ~6.8 MB
// ---------------------------------------------------------------------------
extern "C" void kernel_launch(void* const* d_in, const int* in_sizes, int n_in,
                              void* d_out, int out_size, void* d_ws, size_t ws_size,
                              hipStream_t stream) {
  const float* x     = (const float*)d_in[0];
  const float* W_ih  = (const float*)d_in[1];
  const float* W_hh  = (const float*)d_in[2];
  const float* b_ih  = (const float*)d_in[3];
  const float* b_hh  = (const float*)d_in[4];
  const float* W_mu  = (const float*)d_in[5];
  const float* b_mu  = (const float*)d_in[6];
  const float* W_sig = (const float*)d_in[7];
  const float* b_sig = (const float*)d_in[8];
  float* out = (float*)d_out;

  float* ws  = (float*)d_ws;
  float* xg  = ws;
  float* hws = ws + 1310720;
  float* ac  = ws + 1638400;
  float* sg  = ws + 1671168;
  float* muw = ws + 1687552;

  xg_gemm<<<BS / 128, 256, 0, stream>>>(x, W_ih, b_ih, b_hh, xg);
  lstm_scan<<<BB, 32, 0, stream>>>(xg, W_hh, hws);
  heads<<<BS / 256, 256, 0, stream>>>(hws, W_mu, b_mu, W_sig, b_sig, ac, sg);
  mu_scan<<<1, 512, 0, stream>>>(ac, muw);
  attn_context<<<dim3(SS / MROWS, BB), 256, 0, stream>>>(x, muw, sg, out);
}